// EfficientGATLayer_59081570124185
// MI455X (gfx1250) — compile-verified
//
#include <hip/hip_runtime.h>
#include <hip/hip_bf16.h>

#define NN 50000
#define EE 600000
#define INC 128
#define OUTC 128
#define HEADS 8
#define HEADC 16

typedef __attribute__((ext_vector_type(2))) float v2f;
typedef __attribute__((ext_vector_type(8))) float v8f;

// ---------------------------------------------------------------------------
// Kernel 1: emb = X @ W^T via V_WMMA_F32_16X16X4_F32.
// One wave per 16x16 output tile. 3125 row tiles x 8 col tiles = 25000 waves.
// A (16x4 f32): lane L holds M=L%16, K = {2*(L>=16), 2*(L>=16)+1}  -> b64 load
// B (4x16 f32): lane L holds N=L%16, same K pair; B[k][n] = W[n][k] -> b64 load
// C/D (16x16 f32): VGPR v holds M = v + 8*(L>=16), N = L%16
// ---------------------------------------------------------------------------
__global__ void gat_gemm_wmma(const float* __restrict__ X,
                              const float* __restrict__ W,
                              float* __restrict__ emb) {
    const int wave  = (blockIdx.x * blockDim.x + threadIdx.x) >> 5;
    const int lane  = threadIdx.x & 31;
    const int tileM = wave >> 3;          // 0..3124
    const int tileN = wave & 7;           // 0..7
    const int half  = lane >> 4;          // 0 or 1  (K offset 0 / 2)
    const int l     = lane & 15;          // M for A, N for B

    const float* __restrict__ arow = X + (size_t)(tileM * 16 + l) * INC + half * 2;
    const float* __restrict__ brow = W + (size_t)(tileN * 16 + l) * INC + half * 2;

    v8f c = {};
#pragma unroll
    for (int kk = 0; kk < INC; kk += 4) {
        v2f a = *(const v2f*)(arow + kk);
        v2f b = *(const v2f*)(brow + kk);
        c = __builtin_amdgcn_wmma_f32_16x16x4_f32(
                /*neg_a=*/false, a, /*neg_b=*/false, b,
                /*c_mod=*/(short)0, c, /*reuse_a=*/false, /*reuse_b=*/false);
    }

    float* __restrict__ op =
        emb + (size_t)(tileM * 16 + half * 8) * OUTC + tileN * 16 + l;
#pragma unroll
    for (int v = 0; v < 8; ++v) op[(size_t)v * OUTC] = c[v];
}

// ---------------------------------------------------------------------------
// Kernel 2: per-node attention scores (a_left/a_right layout: [c][h] -> c*8+h)
// ---------------------------------------------------------------------------
__global__ void gat_scores(const float* __restrict__ emb,
                           const float* __restrict__ a_left,
                           const float* __restrict__ a_right,
                           float* __restrict__ ss, float* __restrict__ ts) {
    int gid = blockIdx.x * blockDim.x + threadIdx.x;
    if (gid >= NN * HEADS) return;
    int n = gid >> 3, h = gid & 7;
    const float* ep = emb + (size_t)n * OUTC + h * HEADC;
    float s0 = 0.f, s1 = 0.f;
#pragma unroll
    for (int c = 0; c < HEADC; ++c) {
        float e = ep[c];
        s0 += e * a_left[c * HEADS + h];
        s1 += e * a_right[c * HEADS + h];
    }
    ss[gid] = s0;
    ts[gid] = s1;
}

// ---------------------------------------------------------------------------
// Kernel 3: init out = bias (broadcast) and denom = 0   (deterministic reset)
// ---------------------------------------------------------------------------
__global__ void gat_init(float* __restrict__ out, const float* __restrict__ bias,
                         float* __restrict__ denom) {
    int gid = blockIdx.x * blockDim.x + threadIdx.x;
    if (gid < NN * OUTC) out[gid] = bias[gid & (OUTC - 1)];
    if (gid < NN * HEADS) denom[gid] = 0.f;
}

// ---------------------------------------------------------------------------
// Kernel 4: per-edge exp(LeakyReLU(score)) + softmax denominator scatter-add
// ---------------------------------------------------------------------------
__global__ void gat_edge_exp(const int* __restrict__ src, const int* __restrict__ trg,
                             const float* __restrict__ ss, const float* __restrict__ ts,
                             float* __restrict__ exps, float* __restrict__ denom) {
    int gid = blockIdx.x * blockDim.x + threadIdx.x;
    if (gid >= EE * HEADS) return;
    int e = gid >> 3, h = gid & 7;
    int s = src[e], t = trg[e];
    float v = ss[s * HEADS + h] + ts[t * HEADS + h];
    v = v > 0.f ? v : 0.2f * v;                 // LeakyReLU(0.2)
    float ex = __expf(v);
    exps[gid] = ex;
    atomicAdd(&denom[t * HEADS + h], ex);
}

// ---------------------------------------------------------------------------
// Kernel 5: weighted aggregation  out[trg, k] += emb[src, k] * attn(e, k/16)
// emb and out are L2-resident (25.6 MB each vs 192 MB L2) -> L2 gathers/atomics
// ---------------------------------------------------------------------------
__global__ void gat_aggregate(const int* __restrict__ src, const int* __restrict__ trg,
                              const float* __restrict__ emb,
                              const float* __restrict__ exps,
                              const float* __restrict__ denom,
                              float* __restrict__ out) {
    long long gid = (long long)blockIdx.x * blockDim.x + threadIdx.x;
    if (gid >= (long long)EE * OUTC) return;
    int e = (int)(gid >> 7);
    int k = (int)(gid & (OUTC - 1));
    int h = k >> 4;
    int s = src[e], t = trg[e];
    float attn = exps[e * HEADS + h] / (denom[t * HEADS + h] + 1e-16f);
    atomicAdd(&out[(size_t)t * OUTC + k], emb[(size_t)s * OUTC + k] * attn);
}

extern "C" void kernel_launch(void* const* d_in, const int* in_sizes, int n_in,
                              void* d_out, int out_size, void* d_ws, size_t ws_size,
                              hipStream_t stream) {
    const float* X      = (const float*)d_in[0];   // [N,128]
    const int*   eidx   = (const int*)d_in[1];     // [2,E]
    const float* W      = (const float*)d_in[2];   // [128,128]
    const float* a_l    = (const float*)d_in[3];   // [16,8,1]
    const float* a_r    = (const float*)d_in[4];   // [16,8,1]
    const float* bias   = (const float*)d_in[5];   // [128]
    float* out = (float*)d_out;                    // [N,128]

    const int* src = eidx;        // edge_index[0]
    const int* trg = eidx + EE;   // edge_index[1]

    // workspace layout (floats)
    float* ws    = (float*)d_ws;
    float* emb   = ws;                                   // N*128 = 6,400,000
    float* ss    = emb + (size_t)NN * OUTC;              // N*8
    float* ts    = ss + (size_t)NN * HEADS;              // N*8
    float* denom = ts + (size_t)NN * HEADS;              // N*8
    float* exps  = denom + (size_t)NN * HEADS;           // E*8

    // 1) GEMM: 25000 waves = 3125 blocks of 256 threads (8 waves each), exact fit
    gat_gemm_wmma<<<3125, 256, 0, stream>>>(X, W, emb);

    // 2) per-node scores
    gat_scores<<<(NN * HEADS + 255) / 256, 256, 0, stream>>>(emb, a_l, a_r, ss, ts);

    // 3) init out with bias, zero denominators
    gat_init<<<(NN * OUTC + 255) / 256, 256, 0, stream>>>(out, bias, denom);

    // 4) edge exp + denominator scatter-add
    gat_edge_exp<<<(EE * HEADS + 255) / 256, 256, 0, stream>>>(src, trg, ss, ts, exps, denom);

    // 5) weighted aggregation (E*128 threads)
    long long nagg = (long long)EE * OUTC;
    gat_aggregate<<<(unsigned)((nagg + 255) / 256), 256, 0, stream>>>(
        src, trg, emb, exps, denom, out);
}